// MRPCEN_42949673608
// MI455X (gfx1250) — compile-verified
//
#include <hip/hip_runtime.h>
#include <hip/hip_bf16.h>
#include <math.h>

// Multi-rate PCEN for MI455X (gfx1250).
//  - HBM-bound (~168 MB / 23.3 TB/s ~= 7.2 us floor); no GEMM structure -> no WMMA.
//  - CDNA5 paths: GLOBAL_LOAD_ASYNC_TO_LDS_B128 (+ s_wait_asynccnt),
//    wave32 ds_bpermute affine scan for the IIR carry propagation,
//    non-temporal b128 stores for the write-once 134 MB output stream.
//  - Pointwise math simplified: log_eps + log1p(m/EPS) == log(m + EPS),
//    so smooth = (m+EPS)^(-alpha) -> 2 logs + 2 exps per element.

#define NS       4
#define B_DIM    16
#define F_DIM    128
#define T_DIM    4096
#define THREADS  256
#define CHUNK    16          // timesteps per thread (T / THREADS)
#define EPS_F    1.0e-5f

typedef float v4f __attribute__((ext_vector_type(4)));   // native vector for NT stores

// ---- CDNA5 helpers -------------------------------------------------------

// Low 32 bits of a generic LDS pointer ARE the wave-relative LDS byte offset
// (ISA 10.2).
__device__ __forceinline__ unsigned lds_off_u32(const void* p) {
    return (unsigned)(unsigned long long)p;
}

// Async copy 16 bytes global -> LDS (GV mode, tracked by ASYNCcnt).
__device__ __forceinline__ void async_ld_b128(unsigned lds_byte_off, const void* gaddr) {
    asm volatile("global_load_async_to_lds_b128 %0, %1, off"
                 :: "v"(lds_byte_off), "v"(gaddr)
                 : "memory");
}

__device__ __forceinline__ void wait_asynccnt0() {
    asm volatile("s_wait_asynccnt 0x0" ::: "memory");
}

// wave32 gather: dst[lane] = src[srcLane & 31]
__device__ __forceinline__ float bperm(float v, int srcLane) {
    return __int_as_float(
        __builtin_amdgcn_ds_bpermute(srcLane << 2, __float_as_int(v)));
}

// ---- kernel --------------------------------------------------------------

__global__ void __launch_bounds__(THREADS)
mrpcen_kernel(const float* __restrict__ x,
              const float* __restrict__ alpha_log,
              const float* __restrict__ delta_log,
              const float* __restrict__ r_log,
              float* __restrict__ out)
{
    // Row buffer: 256 chunks, each 4 data float4 + 1 pad float4 (bank decorrelation,
    // keeps every async B128 destination 16B-aligned). 20.5 KB LDS total.
    __shared__ float4 sx4[THREADS * 5];
    __shared__ float  wsum[THREADS / 32];

    const int row  = blockIdx.x;          // 0 .. B*F-1
    const int b    = row / F_DIM;
    const int f    = row - b * F_DIM;
    const int tid  = threadIdx.x;
    const int lane = tid & 31;
    const int wave = tid >> 5;

    const float* xrow = x + (size_t)row * T_DIM;

    // ---- stage the whole row (16 KB) into LDS via the async pipe ----
    #pragma unroll
    for (int i = 0; i < 4; ++i) {
        const int g   = i * THREADS + tid;          // float4 group id, 0..1023
        const int c   = g >> 2;                     // chunk (thread) id
        const int sub = g & 3;                      // float4 within chunk
        async_ld_b128(lds_off_u32(&sx4[c * 5 + sub]), xrow + (size_t)g * 4);
    }
    wait_asynccnt0();
    __syncthreads();

    // ---- per-frequency parameters (delta^r = exp(r*delta_log) folded) ----
    const float alpha = __expf(alpha_log[f]);
    const float r     = __expf(r_log[f]);
    const float delta = __expf(delta_log[f]);
    const float dr    = __expf(r * delta_log[f]);

    // ---- pull this thread's 16 timesteps into registers (read LDS once) ----
    float xv[CHUNK];
    #pragma unroll
    for (int k = 0; k < 4; ++k) {
        const float4 q = sx4[tid * 5 + k];
        xv[4 * k + 0] = q.x; xv[4 * k + 1] = q.y;
        xv[4 * k + 2] = q.z; xv[4 * k + 3] = q.w;
    }
    const float x0 = sx4[0].x;            // broadcast of x[row, 0]

    const float tvals[NS] = {0.015f, 0.06f, 0.25f, 1.0f};

    #pragma unroll
    for (int si = 0; si < NS; ++si) {
        // s = (sqrt(1+4t^2)-1)/(2t^2), t = T * 44100/512  (constant-folds)
        const float tv = tvals[si] * (44100.0f / 512.0f);
        const float s  = (sqrtf(1.0f + 4.0f * tv * tv) - 1.0f) / (2.0f * tv * tv);
        const float P  = 1.0f - s;

        // ---- pass 1: local affine part C = sum_j s*P^(15-j)*xv[j] ----
        float c = 0.0f;
        #pragma unroll
        for (int j = 0; j < CHUNK; ++j) c = fmaf(P, c, s * xv[j]);

        // per-thread decay A = P^16 and its powers
        float A = P * P; A = A * A; A = A * A; A = A * A;      // P^16
        const float A1 = A, A2 = A1 * A1, A4 = A2 * A2;
        const float A8 = A4 * A4, A16 = A8 * A8, A32 = A16 * A16;
        const float A64 = A32 * A32, A128 = A64 * A64;

        // ---- wave32 weighted Kogge-Stone inclusive scan of C ----
        float y = c, u;
        u = bperm(y, lane - 1);  if (lane >= 1)  y = fmaf(A1,  u, y);
        u = bperm(y, lane - 2);  if (lane >= 2)  y = fmaf(A2,  u, y);
        u = bperm(y, lane - 4);  if (lane >= 4)  y = fmaf(A4,  u, y);
        u = bperm(y, lane - 8);  if (lane >= 8)  y = fmaf(A8,  u, y);
        u = bperm(y, lane - 16); if (lane >= 16) y = fmaf(A16, u, y);
        const float e = (lane == 0) ? 0.0f : bperm(y, lane - 1);  // exclusive

        // ---- cross-wave combine (8 partials through LDS) ----
        if (lane == 31) wsum[wave] = y;
        __syncthreads();
        float Wp = 0.0f;                    // carry entering this wave
        for (int w2 = 0; w2 < wave; ++w2) Wp = fmaf(A32, Wp, wsum[w2]);
        __syncthreads();                    // wsum reused next si

        // A^lane, (A^32)^wave by binary decomposition (exact, 8 mults)
        float Al = 1.0f;
        if (lane & 1)  Al *= A1;
        if (lane & 2)  Al *= A2;
        if (lane & 4)  Al *= A4;
        if (lane & 8)  Al *= A8;
        if (lane & 16) Al *= A16;
        float Aw = 1.0f;
        if (wave & 1) Aw *= A32;
        if (wave & 2) Aw *= A64;
        if (wave & 4) Aw *= A128;

        // carry into this thread's chunk; virtual m[-1] = x0 makes m[0] = x0 exact
        float m = fmaf(Aw * Al, x0, fmaf(Al, Wp, e));

        // ---- pass 2: recompute smoother + PCEN pointwise, NT b128 stores ----
        // smooth = (m+EPS)^(-alpha); out = (x*smooth+delta)^r - delta^r
        float* obase = out + ((((size_t)b * NS + si) * F_DIM + f) * T_DIM)
                           + (size_t)tid * CHUNK;
        #pragma unroll
        for (int k = 0; k < 4; ++k) {
            float o[4];
            #pragma unroll
            for (int jj = 0; jj < 4; ++jj) {
                const int j = 4 * k + jj;
                m = fmaf(P, m, s * xv[j]);
                const float sm   = __expf(-alpha * __logf(m + EPS_F));
                const float base = fmaf(xv[j], sm, delta);
                o[jj] = __expf(r * __logf(base)) - dr;
            }
            v4f q;
            q.x = o[0]; q.y = o[1]; q.z = o[2]; q.w = o[3];
            __builtin_nontemporal_store(q, (v4f*)obase + k);
        }
    }
}

// ---- launch --------------------------------------------------------------

extern "C" void kernel_launch(void* const* d_in, const int* in_sizes, int n_in,
                              void* d_out, int out_size, void* d_ws, size_t ws_size,
                              hipStream_t stream) {
    const float* x  = (const float*)d_in[0];
    const float* al = (const float*)d_in[1];
    const float* dl = (const float*)d_in[2];
    const float* rl = (const float*)d_in[3];
    float* out = (float*)d_out;

    dim3 grid(B_DIM * F_DIM);   // 2048 workgroups, one (b,f) row each
    dim3 block(THREADS);        // 8 wave32 per workgroup
    mrpcen_kernel<<<grid, block, 0, stream>>>(x, al, dl, rl, out);
}